// TestNN_68667937128842
// MI455X (gfx1250) — compile-verified
//
#include <hip/hip_runtime.h>

// Problem constants (match the JAX reference)
#define GROUPS   4
#define SSIZE    16
#define NLAYERS  16
#define DDIM     1024
#define BATCH    256
#define D_LOC    256
#define NBUCKETS (GROUPS * SSIZE)   // 64
#define KC       512                // K-chunk staged in LDS per pass
#define STR      516                // LDS row stride in floats (516 % 64 == 4 -> conflict-free b64)

typedef float v2f __attribute__((ext_vector_type(2)));
typedef float v8f __attribute__((ext_vector_type(8)));

// ---------------------------------------------------------------------------
// Bucketing: selector is layer-invariant, so build (group, slot) -> sample
// lists once. counts[g*S+s], idxlist[(g*S+s)*BATCH + i].
// ---------------------------------------------------------------------------
__global__ void zero_counts_kernel(int* __restrict__ counts) {
    if (threadIdx.x < NBUCKETS) counts[threadIdx.x] = 0;
}

__global__ void build_buckets_kernel(const int* __restrict__ selector,
                                     int* __restrict__ counts,
                                     int* __restrict__ idxlist) {
    const int b = threadIdx.x;              // BATCH threads
    #pragma unroll
    for (int g = 0; g < GROUPS; ++g) {
        const int s      = selector[b * GROUPS + g];
        const int bucket = g * SSIZE + s;
        const int pos    = atomicAdd(&counts[bucket], 1);
        idxlist[bucket * BATCH + pos] = b;
    }
}

// ---------------------------------------------------------------------------
// One layer: per-bucket GEMM  (rows = samples selecting this slot,
// N = 256 output cols, K = 1024) using V_WMMA_F32_16X16X4_F32.
// grid = (64 buckets, 4 col-blocks of 64), block = 128 threads (4 waves).
// ---------------------------------------------------------------------------
__global__ __launch_bounds__(128)
void layer_kernel(const float* __restrict__ xin,        // (BATCH, DDIM)
                  const float* __restrict__ storage_l,  // (G, S, DDIM, D_LOC) for this layer
                  const int*   __restrict__ counts,
                  const int*   __restrict__ idxlist,
                  float*       __restrict__ xout)       // (BATCH, DDIM)
{
    __shared__ float xtile[16 * STR];     // 16 rows x 512-float K-chunk (+pad) = ~33 KB

    const int bucket = blockIdx.x;        // 0..63
    const int g      = bucket / SSIZE;
    const int s      = bucket % SSIZE;
    const int n      = counts[bucket];
    if (n == 0) return;

    const int lane = threadIdx.x & 31;
    const int wave = threadIdx.x >> 5;    // 0..3
    const int hi   = lane >> 4;           // half-wave: 0 or 1
    const int l16  = lane & 15;

    // Weight slice for this (g, s): row-major (DDIM x D_LOC)
    const float* Wbase = storage_l + (size_t)(g * SSIZE + s) * (DDIM * D_LOC);
    const int    col   = blockIdx.y * 64 + wave * 16 + l16;       // 0..255
    // B-operand lane pointer: half-wave 1 reads K-rows k0+2 / k0+3
    const float* Wl = Wbase + (size_t)(2 * hi) * D_LOC + col;

    const int* blist = idxlist + bucket * BATCH;

    for (int r0 = 0; r0 < n; r0 += 16) {
        v8f acc = {};                      // 16x16 fp32 accumulator (8 VGPRs)

        for (int kc = 0; kc < DDIM; kc += KC) {
            // Cooperative gather of 16 activation rows (this K-chunk) into LDS.
            // Padded rows are zero so they contribute nothing (stores masked too).
            for (int e = threadIdx.x; e < 16 * (KC / 4); e += blockDim.x) {
                const int r  = e >> 7;          // 0..15
                const int c4 = e & 127;         // float4 index within chunk
                float4 v = make_float4(0.f, 0.f, 0.f, 0.f);
                if (r0 + r < n) {
                    const int bidx = blist[r0 + r];
                    v = *(const float4*)(xin + (size_t)bidx * DDIM + kc + c4 * 4);
                }
                float* dst = &xtile[r * STR + c4 * 4];
                dst[0] = v.x; dst[1] = v.y; dst[2] = v.z; dst[3] = v.w;
            }
            __syncthreads();

            // A-operand lane pointer per ISA 16x4 f32 layout:
            // lane<16: (M=lane, K=k0,k0+1); lane>=16: (M=lane-16, K=k0+2,k0+3)
            const float* xrow = &xtile[l16 * STR + 2 * hi];

            #pragma unroll 8
            for (int k0 = 0; k0 < KC; k0 += 4) {
                const v2f a = *(const v2f*)(xrow + k0);          // ds_load_b64
                v2f bm;
                bm.x = Wl[(size_t)(kc + k0)     * D_LOC];        // K row, coalesced b32
                bm.y = Wl[(size_t)(kc + k0 + 1) * D_LOC];        // K+1 row
                acc = __builtin_amdgcn_wmma_f32_16x16x4_f32(
                        /*neg_a=*/false, a, /*neg_b=*/false, bm,
                        /*c_mod=*/(short)0, acc,
                        /*reuse_a=*/false, /*reuse_b=*/false);
            }
            __syncthreads();
        }

        // ReLU + scatter per C/D layout: VGPR j -> (M = j + 8*hi, N = l16)
        #pragma unroll
        for (int j = 0; j < 8; ++j) {
            const int m = r0 + j + 8 * hi;
            if (m < n) {
                const int bidx = blist[m];
                const float v  = acc[j];
                xout[(size_t)bidx * DDIM + g * D_LOC + col] = v > 0.f ? v : 0.f;
            }
        }
    }
}

// ---------------------------------------------------------------------------
// Host launcher. Inputs: d_in[0]=selector (int32, BATCH*GROUPS),
// d_in[1]=inputs (f32, BATCH*DDIM), d_in[2]=storage (f32, L*G*S*DDIM*D_LOC).
// d_out: BATCH*DDIM f32. Workspace: counts | idxlist | 2 activation buffers.
// ---------------------------------------------------------------------------
extern "C" void kernel_launch(void* const* d_in, const int* in_sizes, int n_in,
                              void* d_out, int out_size, void* d_ws, size_t ws_size,
                              hipStream_t stream) {
    (void)in_sizes; (void)n_in; (void)out_size; (void)ws_size;

    const int*   selector = (const int*)  d_in[0];
    const float* inputs   = (const float*)d_in[1];
    const float* storage  = (const float*)d_in[2];
    float*       out      = (float*)d_out;

    int*   counts  = (int*)d_ws;                       // 64 ints
    int*   idxlist = counts + NBUCKETS;                // 64*256 ints
    float* xbuf0   = (float*)(idxlist + NBUCKETS * BATCH);
    float* xbuf1   = xbuf0 + (size_t)BATCH * DDIM;

    zero_counts_kernel<<<1, 64, 0, stream>>>(counts);
    build_buckets_kernel<<<1, BATCH, 0, stream>>>(selector, counts, idxlist);

    const size_t layer_elems = (size_t)GROUPS * SSIZE * DDIM * D_LOC;
    dim3 grid(NBUCKETS, 4);   // 64 buckets x 4 column blocks (64 cols each)
    dim3 block(128);

    for (int l = 0; l < NLAYERS; ++l) {
        const float* xin  = (l == 0) ? inputs : ((l & 1) ? xbuf0 : xbuf1);
        float*       xout = (l == NLAYERS - 1) ? out : ((l & 1) ? xbuf1 : xbuf0);
        layer_kernel<<<grid, block, 0, stream>>>(
            xin, storage + (size_t)l * layer_elems, counts, idxlist, xout);
    }
}